// MPC_71476845740571
// MI455X (gfx1250) — compile-verified
//
#include <hip/hip_runtime.h>

// Batched iLQR (MPC) for MI455X / gfx1250.
// One wave32 owns one batch element end-to-end:
//   3 x { backward Riccati (T=50, WMMA f32 16x16x4 chains), 5-alpha line-search
//         rollouts, best-alpha re-rollout writing taus / u_nom }.
// Per-step C (1KB) and F (768B) tiles are double-buffered in LDS via
// GLOBAL_LOAD_ASYNC_TO_LDS_B128 (ASYNCcnt), so the DMA of tile t+/-1 overlaps
// the compute of tile t inside the serial time recursion.

#define NS  12
#define NC  4
#define NSC 16
#define T_STEPS 50
#define BATCH   2048
#define LQR_IT  3
#define N_LS    5
#define WAVES_PER_BLOCK 8

typedef __attribute__((ext_vector_type(2))) float v2f;
typedef __attribute__((ext_vector_type(8))) float v8f;

struct WaveLds {
    __align__(16) float Cb[2][256];  // staged C tile, double buffered
    __align__(16) float Fb[2][256];  // staged F tile (rows 12..15 stay zero)
    __align__(16) float V[256];      // padded V, only [0..11][0..11] nonzero
    __align__(16) float Q[256];      // Q = C + F^T V F
    __align__(16) float W[256];      // scratch: V*F result, later QuuK rows 0..3
    float q[16];
    float vv[16];
    float tmp[16];
    float K[64];                     // K[u*16 + j], u<4, j<12
    float kk[4];
    float tau[16];
    float x[16];
    float red[32];
};

// --- async global->LDS tile staging (gfx1250, ASYNCcnt) ---------------------
// INST_OFFSET is added to BOTH the global and LDS addresses, so one base
// address pair plus offset:512 covers each tile's second half.
__device__ __forceinline__ void stage_tile_async(const float* Ct, const float* Ft,
                                                 float* Cb, float* Fb, int lane) {
    unsigned cb = (unsigned)(uintptr_t)Cb + (unsigned)lane * 16u;
    unsigned fb = (unsigned)(uintptr_t)Fb + (unsigned)lane * 16u;
    const float* cg = Ct + lane * 4;
    const float* fg = Ft + lane * 4;
    asm volatile("global_load_async_to_lds_b128 %0, %1, off"
                 :: "v"(cb), "v"(cg) : "memory");
    asm volatile("global_load_async_to_lds_b128 %0, %1, off offset:512"
                 :: "v"(cb), "v"(cg) : "memory");
    asm volatile("global_load_async_to_lds_b128 %0, %1, off"
                 :: "v"(fb), "v"(fg) : "memory");
    if (lane < 16) {   // last 256B of the 768B F tile
        asm volatile("global_load_async_to_lds_b128 %0, %1, off offset:512"
                     :: "v"(fb), "v"(fg) : "memory");
    }
}
__device__ __forceinline__ void wait_async() {
    asm volatile("s_wait_asynccnt 0x0" ::: "memory");
}

// --- WMMA f32 16x16x4 fragment helpers (ISA 7.12.2 layouts) -----------------
__device__ __forceinline__ v2f load_a(const float* M, int c0, int lane) {
    int m = lane & 15, khi = (lane >> 4) << 1;
    v2f a; a.x = M[m * 16 + c0 + khi]; a.y = M[m * 16 + c0 + khi + 1]; return a;
}
__device__ __forceinline__ v2f load_aT(const float* M, int r0, int lane) {
    int m = lane & 15, khi = (lane >> 4) << 1;
    v2f a; a.x = M[(r0 + khi) * 16 + m]; a.y = M[(r0 + khi + 1) * 16 + m]; return a;
}
__device__ __forceinline__ v2f load_bf(const float* M, int r0, int lane) {
    int n = lane & 15, khi = (lane >> 4) << 1;
    v2f b; b.x = M[(r0 + khi) * 16 + n]; b.y = M[(r0 + khi + 1) * 16 + n]; return b;
}
__device__ __forceinline__ void store_cd(float* M, v8f d, int lane) {
    int n = lane & 15, mb = (lane >> 4) << 3;
#pragma unroll
    for (int i = 0; i < 8; ++i) M[(mb + i) * 16 + n] = d[i];
}

// --- forward rollout (double-buffered async staging) -------------------------
__device__ __forceinline__ float do_rollout(
    float alpha, bool write_unom, bool write_out, int b, int lane, WaveLds& L,
    const float* __restrict__ x0, const float* __restrict__ Cg,
    const float* __restrict__ cg, const float* __restrict__ Fg,
    const float* __restrict__ fg, const float* __restrict__ Kws,
    const float* __restrict__ kkws, float* __restrict__ unom,
    float* __restrict__ out)
{
    if (lane < NS) L.x[lane] = x0[(size_t)b * NS + lane];
    // prologue: stage t=0
    stage_tile_async(Cg + (size_t)b * 256, Fg + (size_t)b * 192,
                     L.Cb[0], L.Fb[0], lane);
    float cost = 0.f;
    for (int t = 0; t < T_STEPS; ++t) {
        const size_t tb = (size_t)t * BATCH + b;
        const float* ct = cg + tb * 16;
        const float* ft = fg + tb * 12;
        const float* Kt = Kws + tb * 48;
        const float* kt = kkws + tb * 4;
        float* ut = unom + tb * 4;
        const int cur = t & 1;

        wait_async();                       // tile t is in LDS
        if (t + 1 < T_STEPS) {              // overlap DMA of tile t+1
            const size_t tb1 = tb + BATCH;
            stage_tile_async(Cg + tb1 * 256, Fg + tb1 * 192,
                             L.Cb[cur ^ 1], L.Fb[cur ^ 1], lane);
        }
        const float* Cl = L.Cb[cur];
        const float* Fl = L.Fb[cur];

        // tau = [x ; clip((1-a)*u_nom + a*(K x + k))]
        if (lane < NS) {
            L.tau[lane] = L.x[lane];
        } else if (lane < NSC) {
            int r = lane - NS;
            float ul = kt[r];
#pragma unroll
            for (int j = 0; j < NS; ++j) ul += Kt[r * NS + j] * L.x[j];
            float u = (1.0f - alpha) * ut[r] + alpha * ul;
            u = fminf(1.0f, fmaxf(-1.0f, u));
            L.tau[lane] = u;
        }
        // cost += 0.5 tau^T C tau + c^T tau
        float contrib = 0.f;
        if (lane < NSC) {
            float y = 0.f;
#pragma unroll
            for (int j = 0; j < NSC; ++j) y += Cl[lane * 16 + j] * L.tau[j];
            contrib = (0.5f * y + ct[lane]) * L.tau[lane];
        }
        L.red[lane] = contrib;
        float csum = 0.f;
#pragma unroll
        for (int j = 0; j < NSC; ++j) csum += L.red[j];
        cost += csum;

        if (write_out && lane < NSC) out[tb * 16 + lane] = L.tau[lane];
        if (write_unom && lane >= NS && lane < NSC) ut[lane - NS] = L.tau[lane];

        // x_next = F tau + f
        if (lane < NS) {
            float xn = ft[lane];
#pragma unroll
            for (int j = 0; j < NSC; ++j) xn += Fl[lane * 16 + j] * L.tau[j];
            L.x[lane] = xn;
        }
    }
    return cost;
}

extern "C" __global__ void __launch_bounds__(256)
mpc_ilqr_kernel(const float* __restrict__ x0, const float* __restrict__ Cg,
                const float* __restrict__ cg, const float* __restrict__ Fg,
                const float* __restrict__ fg, float* __restrict__ out,
                float* __restrict__ Kws, float* __restrict__ kkws,
                float* __restrict__ unom)
{
    __shared__ WaveLds lds[WAVES_PER_BLOCK];
    const int lane = threadIdx.x & 31;
    const int wid  = threadIdx.x >> 5;
    const int b    = blockIdx.x * WAVES_PER_BLOCK + wid;
    WaveLds& L = lds[wid];

    for (int i = lane; i < 256; i += 32) {
        L.V[i] = 0.f;
        if (i >= 192) { L.Fb[0][i] = 0.f; L.Fb[1][i] = 0.f; }  // F pad rows
    }
    L.red[lane] = 0.f;
    if (lane < 16) { L.vv[lane] = 0.f; L.x[lane] = 0.f; L.tau[lane] = 0.f; }
    // u_nom = 0 (this wave's batch slice)
    for (int i = lane; i < T_STEPS * NC; i += 32) {
        int t = i >> 2, r = i & 3;
        unom[((size_t)t * BATCH + b) * NC + r] = 0.f;
    }

    for (int iter = 0; iter < LQR_IT; ++iter) {
        // ================= backward Riccati =================
        for (int i = lane; i < 256; i += 32) L.V[i] = 0.f;
        if (lane < 16) L.vv[lane] = 0.f;

        {   // prologue: stage t = T-1
            const size_t tb0 = (size_t)(T_STEPS - 1) * BATCH + b;
            stage_tile_async(Cg + tb0 * 256, Fg + tb0 * 192,
                             L.Cb[(T_STEPS - 1) & 1], L.Fb[(T_STEPS - 1) & 1], lane);
        }
        for (int t = T_STEPS - 1; t >= 0; --t) {
            const size_t tb = (size_t)t * BATCH + b;
            const float* ct = cg + tb * 16;
            const float* ft = fg + tb * 12;
            const int cur = t & 1;

            wait_async();                   // tile t is in LDS
            if (t > 0) {                    // overlap DMA of tile t-1
                const size_t tbm = tb - BATCH;
                stage_tile_async(Cg + tbm * 256, Fg + tbm * 192,
                                 L.Cb[cur ^ 1], L.Fb[cur ^ 1], lane);
            }
            if (t > 1) {                    // warm L2 two tiles ahead
                __builtin_prefetch(Cg + (tb - 2 * (size_t)BATCH) * 256 + lane * 8, 0, 3);
            }
            const float* Cl = L.Cb[cur];
            const float* Fl = L.Fb[cur];

            if (lane < NS) L.red[lane] = ft[lane];   // stage f

            // GEMM1: W = Vpad * Fpad  (chain 4x K=4 WMMAs)
            v8f acc = {};
#pragma unroll
            for (int kb = 0; kb < 4; ++kb) {
                v2f a  = load_a(L.V, 4 * kb, lane);
                v2f bb = load_bf(Fl, 4 * kb, lane);
                acc = __builtin_amdgcn_wmma_f32_16x16x4_f32(
                    false, a, false, bb, (short)0, acc, false, false);
            }
            store_cd(L.W, acc, lane);

            // GEMM2: Q = C + Fpad^T * W  (accumulator seeded from staged C)
            {
                int n = lane & 15, mb = (lane >> 4) << 3;
                v8f qa;
#pragma unroll
                for (int i = 0; i < 8; ++i) qa[i] = Cl[(mb + i) * 16 + n];
#pragma unroll
                for (int kb = 0; kb < 4; ++kb) {
                    v2f a  = load_aT(Fl, 4 * kb, lane);
                    v2f bb = load_bf(L.W, 4 * kb, lane);
                    qa = __builtin_amdgcn_wmma_f32_16x16x4_f32(
                        false, a, false, bb, (short)0, qa, false, false);
                }
                store_cd(L.Q, qa, lane);
            }

            // tmp = V f + v ; q = c + F^T tmp
            if (lane < NS) {
                float s = L.vv[lane];
#pragma unroll
                for (int j = 0; j < NS; ++j) s += L.V[lane * 16 + j] * L.red[j];
                L.tmp[lane] = s;
            }
            if (lane < NSC) {
                float s = ct[lane];
#pragma unroll
                for (int k = 0; k < NS; ++k) s += Fl[k * 16 + lane] * L.tmp[k];
                L.q[lane] = s;
            }

            // Quu * sol = [Qux | qu] ; lane-parallel over 13 RHS columns
            {
                float M4[4][4], r4[4];
#pragma unroll
                for (int i = 0; i < 4; ++i)
#pragma unroll
                    for (int j = 0; j < 4; ++j)
                        M4[i][j] = L.Q[(NS + i) * 16 + (NS + j)];
                int col = (lane < 13) ? lane : 12;
#pragma unroll
                for (int i = 0; i < 4; ++i)
                    r4[i] = (col < NS) ? L.Q[(NS + i) * 16 + col] : L.q[NS + i];
#pragma unroll
                for (int p = 0; p < 4; ++p) {       // Gauss-Jordan (Quu SPD)
                    float inv = 1.0f / M4[p][p];
#pragma unroll
                    for (int j = 0; j < 4; ++j) M4[p][j] *= inv;
                    r4[p] *= inv;
#pragma unroll
                    for (int rr = 0; rr < 4; ++rr) {
                        if (rr != p) {
                            float fch = M4[rr][p];
#pragma unroll
                            for (int j = 0; j < 4; ++j) M4[rr][j] -= fch * M4[p][j];
                            r4[rr] -= fch * r4[p];
                        }
                    }
                }
                if (lane < NS) {
#pragma unroll
                    for (int i = 0; i < 4; ++i) L.K[i * 16 + lane] = -r4[i];
                } else if (lane == 12) {
#pragma unroll
                    for (int i = 0; i < 4; ++i) L.kk[i] = -r4[i];
                }
            }

            // QuuK -> W rows 0..3 (W is free now)
            for (int idx = lane; idx < 48; idx += 32) {
                int u = idx / 12, j = idx % 12;
                float s = 0.f;
#pragma unroll
                for (int w = 0; w < 4; ++w)
                    s += L.Q[(NS + u) * 16 + (NS + w)] * L.K[w * 16 + j];
                L.W[u * 16 + j] = s;
            }
            // g = qu + Quu kk -> red[0..3]
            if (lane < NC) {
                float s = L.q[NS + lane];
#pragma unroll
                for (int w = 0; w < 4; ++w)
                    s += L.Q[(NS + lane) * 16 + (NS + w)] * L.kk[w];
                L.red[lane] = s;
            }
            // V_new (symmetrized), written in place over V pad-safe region
            for (int idx = lane; idx < 144; idx += 32) {
                int i = idx / 12, j = idx % 12;
                float s = 0.5f * (L.Q[i * 16 + j] + L.Q[j * 16 + i]);
                float t1 = 0.f, t2 = 0.f, t3 = 0.f, t4 = 0.f;
#pragma unroll
                for (int u = 0; u < 4; ++u) {
                    t1 += L.Q[i * 16 + NS + u] * L.K[u * 16 + j];
                    t2 += L.Q[j * 16 + NS + u] * L.K[u * 16 + i];
                    t3 += L.K[u * 16 + i] * L.W[u * 16 + j];
                    t4 += L.K[u * 16 + j] * L.W[u * 16 + i];
                }
                L.V[i * 16 + j] = s + t1 + t2 + 0.5f * (t3 + t4);
            }
            // v_new
            if (lane < NS) {
                float s = L.q[lane];
#pragma unroll
                for (int u = 0; u < 4; ++u) {
                    s += L.Q[lane * 16 + NS + u] * L.kk[u];
                    s += L.K[u * 16 + lane] * L.red[u];
                }
                L.vv[lane] = s;
            }
            // persist gains for the forward pass
            for (int idx = lane; idx < 48; idx += 32)
                Kws[tb * 48 + idx] = L.K[(idx / 12) * 16 + (idx % 12)];
            if (lane < NC) kkws[tb * 4 + lane] = L.kk[lane];
        }

        // ================= line search =================
        float bestCost = 3.4e38f;
        int bestA = 0;
        float alpha = 1.0f;
        for (int a = 0; a < N_LS; ++a) {
            float cst = do_rollout(alpha, false, false, b, lane, L,
                                   x0, Cg, cg, Fg, fg, Kws, kkws, unom, out);
            if (cst < bestCost) { bestCost = cst; bestA = a; }
            alpha *= 0.5f;
        }
        float bestAlpha = 1.0f;
        for (int a = 0; a < bestA; ++a) bestAlpha *= 0.5f;
        do_rollout(bestAlpha, true, iter == LQR_IT - 1, b, lane, L,
                   x0, Cg, cg, Fg, fg, Kws, kkws, unom, out);
    }
}

extern "C" void kernel_launch(void* const* d_in, const int* in_sizes, int n_in,
                              void* d_out, int out_size, void* d_ws, size_t ws_size,
                              hipStream_t stream) {
    const float* x0 = (const float*)d_in[0];
    const float* C  = (const float*)d_in[1];
    const float* c  = (const float*)d_in[2];
    const float* F  = (const float*)d_in[3];
    const float* f  = (const float*)d_in[4];
    float* out = (float*)d_out;
    float* ws  = (float*)d_ws;   // needs T*B*56 floats (~23 MB)
    const size_t TB = (size_t)T_STEPS * BATCH;
    float* Kws  = ws;            // [T,B,4,12]
    float* kkws = ws + TB * 48;  // [T,B,4]
    float* unom = ws + TB * 52;  // [T,B,4]
    dim3 grid(BATCH / WAVES_PER_BLOCK);
    dim3 block(WAVES_PER_BLOCK * 32);
    mpc_ilqr_kernel<<<grid, block, 0, stream>>>(x0, C, c, F, f, out,
                                                Kws, kkws, unom);
}